// HybridLossDynamic_49392123904269
// MI455X (gfx1250) — compile-verified
//
#include <hip/hip_runtime.h>

typedef __attribute__((ext_vector_type(2))) float v2f;
typedef __attribute__((ext_vector_type(8))) float v8f;

// ---- normalized 11-tap Gaussian (sigma=1.5), padded by 16 zeros each side
// so index (k - j + 15) for k in [0,27], j in [0,15] is always in range.
__device__ const float GWPAD[43] = {
    0, 0, 0, 0, 0, 0, 0, 0, 0, 0, 0, 0, 0, 0, 0, 0,
    0.00102840f, 0.00759875f, 0.03600077f, 0.10936070f, 0.21300553f,
    0.26601174f, 0.21300553f, 0.10936070f, 0.03600077f, 0.00759875f,
    0.00102840f,
    0, 0, 0, 0, 0, 0, 0, 0, 0, 0, 0, 0, 0, 0, 0, 0};

// ---------------- Sobel base kernels (n1=1, n2=2, n3*n2=4) ----------------
constexpr float SXW[3][3][3] = {
    {{-1, 0, 1}, {-2, 0, 2}, {-1, 0, 1}},
    {{-2, 0, 2}, {-4, 0, 4}, {-2, 0, 2}},
    {{-1, 0, 1}, {-2, 0, 2}, {-1, 0, 1}}};
constexpr float SYW[3][3][3] = {
    {{-1, -2, -1}, {0, 0, 0}, {1, 2, 1}},
    {{-2, -4, -2}, {0, 0, 0}, {2, 4, 2}},
    {{-1, -2, -1}, {0, 0, 0}, {1, 2, 1}}};
constexpr float SZW[3][3][3] = {
    {{-1, -2, -1}, {-2, -4, -2}, {-1, -2, -1}},
    {{0, 0, 0}, {0, 0, 0}, {0, 0, 0}},
    {{1, 2, 1}, {2, 4, 2}, {1, 2, 1}}};
constexpr float SD11[3][3][3] = {
    {{0, 1, 2}, {-1, 0, 1}, {-2, -1, 0}},
    {{0, 2, 4}, {-2, 0, 2}, {-4, -2, 0}},
    {{0, 1, 2}, {-1, 0, 1}, {-2, -1, 0}}};
constexpr float SD12[3][3][3] = {
    {{-2, -1, 0}, {-1, 0, 1}, {0, 1, 2}},
    {{-4, -2, 0}, {-2, 0, 2}, {0, 2, 4}},
    {{-2, -1, 0}, {-1, 0, 1}, {0, 1, 2}}};

__device__ __forceinline__ int refl(int i) {
  return i < 0 ? -i : (i > 127 ? 254 - i : i);
}

// ------------------------------- init ------------------------------------
__global__ void init_acc(double* acc, unsigned* mm) {
  if (threadIdx.x == 0) {
    acc[0] = 0.0;  // ssim sum
    acc[1] = 0.0;  // mse  sum
    acc[2] = 0.0;  // gme  sum
    mm[0] = 0x7f7fffffu; mm[1] = 0u;   // pred  min/max bits
    mm[2] = 0x7f7fffffu; mm[3] = 0u;   // targ  min/max bits
  }
}

// ------------------- build the 5 SSIM fields -----------------------------
__global__ __launch_bounds__(256) void prep_fields(
    const float* __restrict__ p, const float* __restrict__ t,
    float* __restrict__ f, long n) {
  long i = (long)blockIdx.x * 256 + threadIdx.x;
  if (i < n) {
    float a = p[i], b = t[i];
    f[i] = a;
    f[n + i] = b;
    f[2 * n + i] = a * a;
    f[3 * n + i] = b * b;
    f[4 * n + i] = a * b;
  }
}

// ------------- separable 11-tap Gaussian pass via fp32 WMMA ---------------
// One wave computes a 16(rows)x16(axis cols) tile:  D = A(16x28) x B(28x16),
// B[k][j] = g[k-j-1] (banded).  Axis stride / row-inner length passed as
// log2 shifts:  sShift in {0,7,14}, iShift == sShift for this problem.
// addr(row, col) = ((row>>iShift) << (7+sShift)) + (row & ((1<<iShift)-1))
//                  + (col << sShift)
__global__ __launch_bounds__(256) void gauss_pass(
    const float* __restrict__ src, float* __restrict__ dst,
    int sShift, int iShift) {
  __shared__ float tin[8][16][33];   // padded: no bank conflicts
  __shared__ float tout[8][16][17];
  __shared__ float Bmat[28][17];     // shared by all 8 waves of the block

  // ---- build B once per block (unconditional loads from padded table) ----
  for (int idx = threadIdx.x; idx < 28 * 16; idx += 256) {
    int k = idx >> 4, j = idx & 15;
    Bmat[k][j] = GWPAD[k - j + 15];
  }

  const int lane = threadIdx.x & 31;
  const int wv = threadIdx.x >> 5;
  const long tile = (long)blockIdx.x * 8 + wv;
  const int colT = (int)(tile & 7);        // 128/16 = 8 col tiles
  const long r0 = (tile >> 3) * 16;        // first of 16 rows
  const int c0 = colT * 16 - 6;            // first staged input column
  const int half = lane >> 4;              // 0: lanes 0-15, 1: lanes 16-31
  const int l15 = lane & 15;
  const long iMask = (1L << iShift) - 1;

  // ---- stage 16x32 input tile into LDS, coalesced for either layout ----
  if (sShift == 0) {                       // conv along W: rows differ by 128
#pragma unroll
    for (int i = 0; i < 16; i++) {
      int cg = c0 + lane;
      long base = (r0 + i) << 7;           // inner == 1
      tin[wv][i][lane] = (cg >= 0 && cg < 128) ? src[base + cg] : 0.0f;
    }
  } else {                                 // 16 consecutive rows contiguous
#pragma unroll
    for (int i = 0; i < 16; i++) {
      int col = i * 2 + half;
      int cg = c0 + col;
      long row = r0 + l15;
      long base = ((row >> iShift) << (7 + sShift)) + (row & iMask);
      tin[wv][l15][col] =
          (cg >= 0 && cg < 128) ? src[base + ((long)cg << sShift)] : 0.0f;
    }
  }
  __syncthreads();

  // ---- 7 x V_WMMA_F32_16X16X4_F32 : banded conv as matmul ----
  // A 16x4 fp32 layout: lanes 0-15 hold K0(v0),K1(v1); lanes 16-31 K2,K3.
  v8f acc = {};
#pragma unroll
  for (int c = 0; c < 7; c++) {
    int k0 = c * 4 + 2 * half;
    v2f a, b;
    a.x = tin[wv][l15][k0];
    a.y = tin[wv][l15][k0 + 1];
    b.x = Bmat[k0][l15];
    b.y = Bmat[k0 + 1][l15];
    acc = __builtin_amdgcn_wmma_f32_16x16x4_f32(
        /*neg_a=*/false, a, /*neg_b=*/false, b,
        /*c_mod=*/(short)0, acc, /*reuse_a=*/false, /*reuse_b=*/false);
  }

  // ---- store D: VGPR v -> (M = v + 8*half, N = l15) ----
  if (sShift == 0) {
#pragma unroll
    for (int v = 0; v < 8; v++) {
      long row = r0 + v + 8 * half;
      dst[(row << 7) + colT * 16 + l15] = acc[v];
    }
  } else {                                 // transpose through LDS, coalesce
#pragma unroll
    for (int v = 0; v < 8; v++) tout[wv][v + 8 * half][l15] = acc[v];
    __syncthreads();
#pragma unroll
    for (int i = 0; i < 8; i++) {
      int col = i * 2 + half;
      long row = r0 + l15;
      long base = ((row >> iShift) << (7 + sShift)) + (row & iMask);
      dst[base + ((long)(colT * 16 + col) << sShift)] = tout[wv][l15][col];
    }
  }
}

// --------------- fused SSIM map + MSE reduction ---------------------------
__global__ __launch_bounds__(256) void ssim_mse_kernel(
    const float* __restrict__ p, const float* __restrict__ t,
    const float* __restrict__ f, double* acc, long n) {
  const float C1 = 1e-4f, C2 = 9e-4f;
  double ls = 0.0, lm = 0.0;
  for (long i = (long)blockIdx.x * 256 + threadIdx.x; i < n;
       i += (long)gridDim.x * 256) {
    float mp = f[i], mt = f[n + i];
    float ep2 = f[2 * n + i], et2 = f[3 * n + i], ept = f[4 * n + i];
    float mp2 = mp * mp, mt2 = mt * mt, mpt = mp * mt;
    float sp = ep2 - mp2, st = et2 - mt2, spt = ept - mpt;
    float num = (2.0f * mpt + C1) * (2.0f * spt + C2);
    float den = (mp2 + mt2 + C1) * (sp + st + C2);
    ls += (double)(num / den);
    float d = p[i] - t[i];
    lm += (double)(d * d);
  }
  __shared__ double sd[256];
  int tx = threadIdx.x;
  sd[tx] = ls; __syncthreads();
  for (int o = 128; o > 0; o >>= 1) {
    if (tx < o) sd[tx] += sd[tx + o];
    __syncthreads();
  }
  if (tx == 0) atomicAdd(&acc[0], sd[0]);
  __syncthreads();
  sd[tx] = lm; __syncthreads();
  for (int o = 128; o > 0; o >>= 1) {
    if (tx < o) sd[tx] += sd[tx + o];
    __syncthreads();
  }
  if (tx == 0) atomicAdd(&acc[1], sd[0]);
}

// --------- 3D Sobel (9 filters, reflect pad) + magnitude + min/max --------
__global__ __launch_bounds__(256) void sobel_kernel(
    const float* __restrict__ img, float* __restrict__ gm, unsigned* mm) {
  long i = (long)blockIdx.x * 256 + threadIdx.x;
  int x = (int)(i & 127), y = (int)((i >> 7) & 127), z = (int)((i >> 14) & 127);
  long b = i >> 21;
  const float* base = img + (b << 21);
  float s[9] = {0, 0, 0, 0, 0, 0, 0, 0, 0};
#pragma unroll
  for (int dz = 0; dz < 3; dz++) {
    int zz = refl(z + dz - 1);
#pragma unroll
    for (int dy = 0; dy < 3; dy++) {
      int yy = refl(y + dy - 1);
#pragma unroll
      for (int dx = 0; dx < 3; dx++) {
        int xx = refl(x + dx - 1);
        float v = base[((long)zz << 14) + (yy << 7) + xx];
        s[0] += SXW[dz][dy][dx] * v;
        s[1] += SYW[dz][dy][dx] * v;
        s[2] += SZW[dz][dy][dx] * v;
        s[3] += SD11[dz][dy][dx] * v;
        s[4] += SD12[dz][dy][dx] * v;
        s[5] += SD11[dx][dy][dz] * v;   // Sd21 = Sd11.T
        s[6] += SD12[dx][dy][dz] * v;   // Sd22 = Sd12.T
        s[7] -= SD11[dy][dx][dz] * v;   // Sd31
        s[8] += SD12[dy][dx][dz] * v;   // Sd32
      }
    }
  }
  float q = 0.0f;
#pragma unroll
  for (int c = 0; c < 9; c++) {
    float tt = s[c] + 1e-6f;
    q += tt * tt;
  }
  q += 9e-6f;                       // sum of per-channel +1e-6 terms
  float g = sqrtf(q);
  gm[i] = g;

  __shared__ float smin[256], smax[256];
  int tx = threadIdx.x;
  smin[tx] = g; smax[tx] = g; __syncthreads();
  for (int o = 128; o > 0; o >>= 1) {
    if (tx < o) {
      smin[tx] = fminf(smin[tx], smin[tx + o]);
      smax[tx] = fmaxf(smax[tx], smax[tx + o]);
    }
    __syncthreads();
  }
  if (tx == 0) {                    // gm > 0: uint bit order == float order
    atomicMin(&mm[0], __float_as_uint(smin[0]));
    atomicMax(&mm[1], __float_as_uint(smax[0]));
  }
}

// ------------- normalized edge-map L1 difference reduction ----------------
__global__ __launch_bounds__(256) void gme_kernel(
    const float* __restrict__ gp, const float* __restrict__ gt,
    const unsigned* mm, double* acc, long n) {
  float minp = __uint_as_float(mm[0]), maxp = __uint_as_float(mm[1]);
  float mint = __uint_as_float(mm[2]), maxt = __uint_as_float(mm[3]);
  float dp = maxp - minp + 1e-6f, dt = maxt - mint + 1e-6f;
  double s = 0.0;
  for (long i = (long)blockIdx.x * 256 + threadIdx.x; i < n;
       i += (long)gridDim.x * 256) {
    float np = (gp[i] - minp) / dp;
    float nt = (gt[i] - mint) / dt;
    s += (double)fabsf(np - nt);
  }
  __shared__ double sd[256];
  int tx = threadIdx.x;
  sd[tx] = s; __syncthreads();
  for (int o = 128; o > 0; o >>= 1) {
    if (tx < o) sd[tx] += sd[tx + o];
    __syncthreads();
  }
  if (tx == 0) atomicAdd(acc, sd[0]);
}

// ------------------------------ finalize ----------------------------------
__global__ void finalize_kernel(const double* acc, float* out, long n) {
  if (threadIdx.x == 0) {
    double mse = acc[1] / (double)n;
    double ssim_loss = 1.0 - acc[0] / (double)n;
    double gme = 1e-6 + acc[2] / (double)n;
    double total = 1.0 * mse + 1.0 * ssim_loss + 0.7 * gme;
    out[0] = (float)total;
    out[1] = (float)mse;
    out[2] = (float)ssim_loss;
    out[3] = (float)gme;
  }
}

extern "C" void kernel_launch(void* const* d_in, const int* in_sizes, int n_in,
                              void* d_out, int out_size, void* d_ws,
                              size_t ws_size, hipStream_t stream) {
  const float* pred = (const float*)d_in[0];
  const float* targ = (const float*)d_in[1];
  float* out = (float*)d_out;
  const long N = (long)in_sizes[0];        // 2*128^3 = 4,194,304

  float* A = (float*)d_ws;                 // 5N floats (fields / gm reuse)
  float* Bb = A + 5 * N;                   // 5N floats (ping-pong)
  double* acc = (double*)((char*)d_ws + (size_t)10 * N * 4);
  unsigned* mm = (unsigned*)(acc + 3);

  init_acc<<<1, 1, 0, stream>>>(acc, mm);
  prep_fields<<<(unsigned)((N + 255) / 256), 256, 0, stream>>>(pred, targ, A, N);

  const unsigned blocksG = (unsigned)((5 * N / 128) / 16);  // 10240
  gauss_pass<<<blocksG, 256, 0, stream>>>(A, Bb, 0, 0);     // W axis
  gauss_pass<<<blocksG, 256, 0, stream>>>(Bb, A, 7, 7);     // H axis
  gauss_pass<<<blocksG, 256, 0, stream>>>(A, Bb, 14, 14);   // D axis

  ssim_mse_kernel<<<2048, 256, 0, stream>>>(pred, targ, Bb, acc, N);

  // field buffer A no longer needed -> reuse for gradient magnitudes
  sobel_kernel<<<(unsigned)(N / 256), 256, 0, stream>>>(pred, A, mm + 0);
  sobel_kernel<<<(unsigned)(N / 256), 256, 0, stream>>>(targ, A + N, mm + 2);

  gme_kernel<<<2048, 256, 0, stream>>>(A, A + N, mm, acc + 2, N);
  finalize_kernel<<<1, 1, 0, stream>>>(acc, out, N);
}